// KernelEncoderLayer_16277926052055
// MI455X (gfx1250) — compile-verified
//
#include <hip/hip_runtime.h>
#include <hip/hip_bf16.h>

// ---------------------------------------------------------------------------
// Problem constants (from reference)
// ---------------------------------------------------------------------------
#define BB     8
#define NB     1024
#define KK     9
#define CC     32
#define CM     128
#define POSD   2
#define NN     (BB * NB)          // 8192
#define MM     (NB * KK)          // 9216 rows of conv_w per batch
#define CTOT   (CC + POSD)        // 34
#define CPAD   48                 // padded W2 cols (3 x 16)
#define BN_EPS 1e-5f
#define NEG_SLOPE 0.01f
#define EXPC   (-2.8853900817779268f)   // -1 / (2*sigma^2) * log2(e), sigma=0.5

typedef __attribute__((ext_vector_type(2))) float v2f;
typedef __attribute__((ext_vector_type(8))) float v8f;

#if __has_builtin(__builtin_amdgcn_wmma_f32_16x16x4_f32)
#define HAVE_WMMA_F32X4 1
#endif

__device__ __forceinline__ float leaky(float x) {
    return x >= 0.0f ? x : NEG_SLOPE * x;
}

// Raw v_exp_f32 (no overflow fixup).  Argument is always <= 0 here, and
// underflow-to-zero is exactly the Gaussian tail we want.
__device__ __forceinline__ float fast_exp2(float x) {
#if __has_builtin(__builtin_amdgcn_exp2f)
    return __builtin_amdgcn_exp2f(x);
#else
    return exp2f(x);
#endif
}

// D(16x16,f32) += A(16x4,f32) * B(4x16,f32), wave32.
// A layout: lane<16 holds M=lane, K=0 (comp x), K=1 (comp y);
//           lane>=16 holds M=lane-16, K=2 (x), K=3 (y).
// B layout mirrored on N. C/D: VGPR r -> row r (lanes 0-15), row r+8 (lanes 16-31).
__device__ __forceinline__ v8f wmma4(v2f a, v2f b, v8f c) {
#ifdef HAVE_WMMA_F32X4
    return __builtin_amdgcn_wmma_f32_16x16x4_f32(
        false, a, false, b, (short)0, c, false, false);
#else
    const int lane = threadIdx.x & 31;
    const int l15  = lane & 15;
    const int hi   = lane >> 4;
    float a0 = a.x, a1 = a.y, b0 = b.x, b1 = b.y;
#pragma unroll
    for (int r = 0; r < 8; ++r) {
        int M = r + 8 * hi;
        float s = __shfl(a0, M, 32)      * __shfl(b0, l15, 32)
                + __shfl(a1, M, 32)      * __shfl(b1, l15, 32)
                + __shfl(a0, M + 16, 32) * __shfl(b0, l15 + 16, 32)
                + __shfl(a1, M + 16, 32) * __shfl(b1, l15 + 16, 32);
        c[r] += s;
    }
    return c;
#endif
}

// ---------------------------------------------------------------------------
// Workspace layout (float offsets)
// ---------------------------------------------------------------------------
#define OFF_CONVW   0                          // [B*MM][32]
#define OFF_CPOS    (OFF_CONVW + BB*MM*CC)     // [B*MM][4]  (cx, cy, |c|^2, 0)
#define OFF_SAMP    (OFF_CPOS + BB*MM*4)       // [N][32]
#define OFF_X       (OFF_SAMP + NN*CC)         // [N][32]
#define OFF_T       (OFF_X + NN*CC)            // [N][128]
#define OFF_STATS0  (OFF_T + NN*CM)            // 64  (sum[32], sumsq[32])
#define OFF_SC1     (OFF_STATS0 + 64)          // 64  (scale[32], shift[32])
#define OFF_STATS1  (OFF_SC1 + 64)             // 256 (sum[128], sumsq[128])
#define OFF_SC2     (OFF_STATS1 + 256)         // 256
#define OFF_W2P     (OFF_SC2 + 256)            // [128][48]
#define OFF_B2P     (OFF_W2P + CM*CPAD)        // 48

// ---------------------------------------------------------------------------
// K0: zero stats, build padded W2/b2
// ---------------------------------------------------------------------------
__global__ void k0_init(float* ws, const float* W2, const float* b2) {
    int t = threadIdx.x;
    for (int i = t; i < 64; i += 256)  ws[OFF_STATS0 + i] = 0.0f;
    for (int i = t; i < 256; i += 256) ws[OFF_STATS1 + i] = 0.0f;
    for (int i = t; i < CM * CPAD; i += 256) {
        int r = i / CPAD, c = i % CPAD;
        ws[OFF_W2P + i] = (c < CTOT) ? W2[r * CTOT + c] : 0.0f;
    }
    for (int i = t; i < CPAD; i += 256)
        ws[OFF_B2P + i] = (i < CTOT) ? b2[i] : 0.0f;
}

// ---------------------------------------------------------------------------
// K1: conv_w[n*9+k][d] = sum_c weights[n][c]*KW[k][c][d];  cpos4[n*9+k]
// grid: 9*32 blocks x 256 thr; one k per block, KW[k] staged in LDS.
// ---------------------------------------------------------------------------
__global__ void k1_prep(float* ws, const float* positions, const float* weights,
                        const float* kpos, const float* kw) {
    __shared__ float skw[CC * CC];
    int k = blockIdx.x >> 5;                 // 32 blocks per k
    int n = ((blockIdx.x & 31) << 8) + threadIdx.x;
    for (int i = threadIdx.x; i < CC * CC; i += 256)
        skw[i] = kw[k * CC * CC + i];
    __syncthreads();

    float w[CC];
#pragma unroll
    for (int c = 0; c < CC; c += 4) {
        float4 v = *(const float4*)(weights + n * CC + c);
        w[c] = v.x; w[c + 1] = v.y; w[c + 2] = v.z; w[c + 3] = v.w;
    }
    float* cw = ws + OFF_CONVW + (size_t)(n * KK + k) * CC;
#pragma unroll 4
    for (int d = 0; d < CC; ++d) {
        float acc = 0.0f;
#pragma unroll
        for (int c = 0; c < CC; ++c) acc = fmaf(w[c], skw[c * CC + d], acc);
        cw[d] = acc;
    }
    float px = positions[n * 2], py = positions[n * 2 + 1];
    float cx = px + kpos[k * 2], cy = py + kpos[k * 2 + 1];
    float4 cp = make_float4(cx, cy, cx * cx + cy * cy, 0.0f);
    *(float4*)(ws + OFF_CPOS + (size_t)(n * KK + k) * 4) = cp;
}

// ---------------------------------------------------------------------------
// K2: fused Gaussian-kernel GEMM with split-K.
// Block = 64 rows x 32 cols of sampled[b]; 8 waves = 4 row-tiles x 2 K-halves.
// Each wave: 16x32 output (two f32 WMMA accumulators sharing one A tile),
// K-loop over 4608 in steps of 4, Kmat = raw v_exp_f32.  K-halves combined
// through LDS.  grid: 128 blocks x 256 thr.
// ---------------------------------------------------------------------------
__global__ void k2_gauss_gemm(float* ws, const float* positions) {
    __shared__ float partial[4][16][32];     // [row-tile][row][col] (khalf=1)

    int blk = blockIdx.x;
    int b = blk >> 4;                 // 8 batches
    int rg = blk & 15;                // 16 row-groups of 64 rows
    int wave = threadIdx.x >> 5;
    int ntile = wave & 3;             // row tile inside block
    int khalf = wave >> 2;            // K-split half
    int lane = threadIdx.x & 31;
    int l15 = lane & 15;
    int hi = lane >> 4;

    int n_base = rg * 64 + ntile * 16;            // row within batch
    int n_glob = b * NB + n_base + l15;
    float px = positions[n_glob * 2];
    float py = positions[n_glob * 2 + 1];
    float p2 = px * px + py * py;

    const float4* cp4 = (const float4*)(ws + OFF_CPOS) + (size_t)b * MM;
    const float* cw = ws + OFF_CONVW + (size_t)b * MM * CC;

    int m0 = khalf * (MM / 2);
    int m1 = m0 + (MM / 2);

    v8f acc0 = {}, acc1 = {};
#pragma unroll 2
    for (int m = m0; m < m1; m += 4) {
        int mk = m + 2 * hi;
        float4 c0 = cp4[mk];
        float4 c1 = cp4[mk + 1];
        // Unconditional prefetch: over-read lands in cpos region (same ws).
        __builtin_prefetch(&cw[(size_t)(mk + 160) * CC], 0, 1);
        float d2a = p2 + c0.z - 2.0f * fmaf(px, c0.x, py * c0.y);
        float d2b = p2 + c1.z - 2.0f * fmaf(px, c1.x, py * c1.y);
        v2f a;
        a.x = fast_exp2(EXPC * d2a);
        a.y = fast_exp2(EXPC * d2b);
        const float* r0 = &cw[(size_t)mk * CC];
        const float* r1 = &cw[(size_t)(mk + 1) * CC];
        v2f b0; b0.x = r0[l15];      b0.y = r1[l15];
        v2f b1; b1.x = r0[16 + l15]; b1.y = r1[16 + l15];
        acc0 = wmma4(a, b0, acc0);
        acc1 = wmma4(a, b1, acc1);
    }

    if (khalf == 1) {
#pragma unroll
        for (int r = 0; r < 8; ++r) {
            partial[ntile][r + 8 * hi][l15]      = acc0[r];
            partial[ntile][r + 8 * hi][16 + l15] = acc1[r];
        }
    }
    __syncthreads();
    if (khalf == 0) {
        float* samp = ws + OFF_SAMP;
#pragma unroll
        for (int r = 0; r < 8; ++r) {
            int rr = r + 8 * hi;
            int row = b * NB + n_base + rr;
            samp[(size_t)row * CC + l15] =
                acc0[r] + partial[ntile][rr][l15];
            samp[(size_t)row * CC + 16 + l15] =
                acc1[r] + partial[ntile][rr][16 + l15];
        }
    }
}

// ---------------------------------------------------------------------------
// K3: batchnorm-1 stats over leaky(sampled): sum/sumsq per channel (32)
// grid: 64 blocks x 256 thr
// ---------------------------------------------------------------------------
__global__ void k3_bn1_stats(float* ws) {
    int t = threadIdx.x;
    int c = t & 31;
    int rg = t >> 5;                  // 0..7
    const float* samp = ws + OFF_SAMP;
    float s = 0.0f, sq = 0.0f;
#pragma unroll 4
    for (int i = 0; i < 16; ++i) {
        int row = blockIdx.x * 128 + rg + 8 * i;
        float v = leaky(samp[(size_t)row * CC + c]);
        s += v; sq += v * v;
    }
    atomicAdd(&ws[OFF_STATS0 + c], s);
    atomicAdd(&ws[OFF_STATS0 + 32 + c], sq);
}

__global__ void k3b_bn1_finalize(float* ws, const float* gamma, const float* beta) {
    int c = threadIdx.x;              // 32 threads
    float mu = ws[OFF_STATS0 + c] * (1.0f / NN);
    float var = ws[OFF_STATS0 + 32 + c] * (1.0f / NN) - mu * mu;
    float sc = gamma[c] * rsqrtf(var + BN_EPS);
    ws[OFF_SC1 + c] = sc;
    ws[OFF_SC1 + 32 + c] = beta[c] - mu * sc;
}

// ---------------------------------------------------------------------------
// K4: x = bn1(leaky(sampled)) + weights;  t = leaky(x @ W1 + b1);
//     accumulate bn2 stats per column via LDS ds_add_f32.
// grid: 256 blocks x 256 thr (8 waves); block handles 32 rows x 128 cols.
// ---------------------------------------------------------------------------
__global__ void k4_mlp1(float* ws, const float* weights, const float* W1,
                        const float* b1) {
    __shared__ float xs[32 * 33];
    __shared__ float colsum[CM];
    __shared__ float colsq[CM];
    int tid = threadIdx.x;
    int r0 = blockIdx.x * 32;
    const float* samp = ws + OFF_SAMP;
    float* xg = ws + OFF_X;
    float* tw = ws + OFF_T;

    if (tid < CM) { colsum[tid] = 0.0f; colsq[tid] = 0.0f; }

#pragma unroll
    for (int i = 0; i < 4; ++i) {
        int e = tid + i * 256;        // 0..1023
        int rr = e >> 5, c = e & 31;
        int n = r0 + rr;
        float v = leaky(samp[(size_t)n * CC + c]);
        float xv = fmaf(v, ws[OFF_SC1 + c], ws[OFF_SC1 + 32 + c])
                 + weights[(size_t)n * CC + c];
        xs[rr * 33 + c] = xv;
        xg[(size_t)n * CC + c] = xv;
    }
    __syncthreads();

    int wave = tid >> 5;
    int lane = tid & 31;
    int l15 = lane & 15;
    int hi = lane >> 4;
#pragma unroll
    for (int tnum = 0; tnum < 2; ++tnum) {
        int t = wave + tnum * 8;      // 0..15 tiles: tr = t&1, tc = t>>1
        int tr = t & 1;
        int tc = t >> 1;
        v8f acc = {};
#pragma unroll
        for (int k = 0; k < CC; k += 4) {
            int kk = k + 2 * hi;
            v2f a; a.x = xs[(16 * tr + l15) * 33 + kk];
                   a.y = xs[(16 * tr + l15) * 33 + kk + 1];
            v2f bv; bv.x = W1[kk * CM + tc * 16 + l15];
                    bv.y = W1[(kk + 1) * CM + tc * 16 + l15];
            acc = wmma4(a, bv, acc);
        }
        int col = tc * 16 + l15;
        float bb = b1[col];
        float ls = 0.0f, lq = 0.0f;
#pragma unroll
        for (int r = 0; r < 8; ++r) {
            int row = 16 * tr + r + 8 * hi;
            float v = leaky(acc[r] + bb);
            tw[(size_t)(r0 + row) * CM + col] = v;
            ls += v; lq += v * v;
        }
        atomicAdd(&colsum[col], ls);   // ds_add_f32
        atomicAdd(&colsq[col], lq);
    }
    __syncthreads();
    if (tid < CM) {
        atomicAdd(&ws[OFF_STATS1 + tid], colsum[tid]);
        atomicAdd(&ws[OFF_STATS1 + CM + tid], colsq[tid]);
    }
}

__global__ void k4b_bn2_finalize(float* ws, const float* gamma, const float* beta) {
    int c = threadIdx.x;              // 128 threads
    float mu = ws[OFF_STATS1 + c] * (1.0f / NN);
    float var = ws[OFF_STATS1 + CM + c] * (1.0f / NN) - mu * mu;
    float sc = gamma[c] * rsqrtf(var + BN_EPS);
    ws[OFF_SC2 + c] = sc;
    ws[OFF_SC2 + CM + c] = beta[c] - mu * sc;
}

// ---------------------------------------------------------------------------
// K5: mlp = bn2(t) @ W2p + b2p;  out0 = positions + mlp[:, :2];
//     out1 = x + mlp[:, 2:34].
// grid: 512 blocks x 96 thr (3 waves); block = 16 rows x 48 cols.
// ---------------------------------------------------------------------------
__global__ void k5_mlp2(float* ws, const float* positions, float* dout) {
    int n_base = blockIdx.x * 16;
    int wave = threadIdx.x >> 5;      // 0..2 -> column tile
    int lane = threadIdx.x & 31;
    int l15 = lane & 15;
    int hi = lane >> 4;
    const float* tw = ws + OFF_T;
    const float* w2p = ws + OFF_W2P;
    const float* xg = ws + OFF_X;

    v8f acc = {};
#pragma unroll 2
    for (int k = 0; k < CM; k += 4) {
        int kk = k + 2 * hi;
        float t0 = tw[(size_t)(n_base + l15) * CM + kk];
        float t1 = tw[(size_t)(n_base + l15) * CM + kk + 1];
        v2f a; a.x = fmaf(t0, ws[OFF_SC2 + kk], ws[OFF_SC2 + CM + kk]);
               a.y = fmaf(t1, ws[OFF_SC2 + kk + 1], ws[OFF_SC2 + CM + kk + 1]);
        v2f bv; bv.x = w2p[kk * CPAD + wave * 16 + l15];
                bv.y = w2p[(kk + 1) * CPAD + wave * 16 + l15];
        acc = wmma4(a, bv, acc);
    }
    int col = wave * 16 + l15;
    float bb = ws[OFF_B2P + col];
#pragma unroll
    for (int r = 0; r < 8; ++r) {
        int row = n_base + r + 8 * hi;
        float v = acc[r] + bb;
        if (col < POSD) {
            dout[(size_t)row * POSD + col] = positions[(size_t)row * POSD + col] + v;
        } else if (col < CTOT) {
            int c = col - POSD;
            dout[(size_t)NN * POSD + (size_t)row * CC + c] =
                xg[(size_t)row * CC + c] + v;
        }
    }
}

// ---------------------------------------------------------------------------
// Launcher
// ---------------------------------------------------------------------------
extern "C" void kernel_launch(void* const* d_in, const int* in_sizes, int n_in,
                              void* d_out, int out_size, void* d_ws, size_t ws_size,
                              hipStream_t stream) {
    const float* positions = (const float*)d_in[0];
    const float* weights   = (const float*)d_in[1];
    // d_in[2] = batch (int), unused: all groups are size NB
    const float* kpos      = (const float*)d_in[3];
    const float* kw        = (const float*)d_in[4];
    const float* g1        = (const float*)d_in[5];
    const float* be1       = (const float*)d_in[6];
    const float* W1        = (const float*)d_in[7];
    const float* b1        = (const float*)d_in[8];
    const float* g2        = (const float*)d_in[9];
    const float* be2       = (const float*)d_in[10];
    const float* W2        = (const float*)d_in[11];
    const float* b2        = (const float*)d_in[12];
    float* ws = (float*)d_ws;
    float* out = (float*)d_out;

    k0_init<<<1, 256, 0, stream>>>(ws, W2, b2);
    k1_prep<<<KK * 32, 256, 0, stream>>>(ws, positions, weights, kpos, kw);
    k2_gauss_gemm<<<128, 256, 0, stream>>>(ws, positions);
    k3_bn1_stats<<<64, 256, 0, stream>>>(ws);
    k3b_bn1_finalize<<<1, 32, 0, stream>>>(ws, g1, be1);
    k4_mlp1<<<256, 256, 0, stream>>>(ws, weights, W1, b1);
    k4b_bn2_finalize<<<1, 128, 0, stream>>>(ws, g2, be2);
    k5_mlp2<<<512, 96, 0, stream>>>(ws, positions, out);
}